// DBLayer_58729382805739
// MI455X (gfx1250) — compile-verified
//
#include <hip/hip_runtime.h>

#define TPB 256
#define KU  4   // async b128 ops per thread per kernel-2 block

typedef float v4f __attribute__((ext_vector_type(4)));

// ---------------------------------------------------------------------------
// CDNA5 async global<->LDS helpers (gfx1250 ASYNCcnt path, CDNA5 ISA)
// ---------------------------------------------------------------------------
__device__ __forceinline__ unsigned lds_off(const void* p) {
    // Flat shared-aperture address: low 32 bits == LDS byte offset (ISA 10.2).
    return (unsigned)(size_t)p;
}

__device__ __forceinline__ void async_ld_b128(unsigned lds, unsigned long long gaddr) {
    // Copies 16 bytes per lane from the global address gaddr to the LDS address lds, tracked by ASYNCcnt.
    // Default RT temporal hint: db is L2-resident (128 MB < 192 MB L2).
    asm volatile("global_load_async_to_lds_b128 %0, %1, off"
                 :: "v"(lds), "v"(gaddr) : "memory");
}

__device__ __forceinline__ void async_st_b128_nt(unsigned long long gaddr, unsigned lds) {
    // Copies 16 bytes per lane from the LDS address lds to the global address gaddr, tracked by ASYNCcnt.
    // NT store: gathered output is write-once; don't evict db from L2.
    asm volatile("global_store_async_from_lds_b128 %0, %1, off th:TH_STORE_NT"
                 :: "v"(gaddr), "v"(lds) : "memory");
}

__device__ __forceinline__ void wait_async0() {
    asm volatile("s_wait_asynccnt 0x0" ::: "memory");
}

// ---------------------------------------------------------------------------
// Kernel 1: prefix sums + per-r start indices + last-write-wins winner table.
// One block of 1024 threads; tables are tiny (8 / 64 / 512 / 1024 entries).
// ---------------------------------------------------------------------------
__global__ void setup_kernel(const int* __restrict__ fields,
                             const int* __restrict__ ppf,
                             const int* __restrict__ sp,
                             const int* __restrict__ coords,
                             int nm, int nf, int P, int R,
                             long long* __restrict__ param_start,
                             long long* __restrict__ starts,
                             int* __restrict__ winner) {
    __shared__ long long s_pf[64];     // exclusive prefix of fields     (nm+1 <= 64)
    __shared__ long long s_pp[256];    // exclusive prefix of ppf        (nf+1 <= 256)
    __shared__ long long s_ps[1024];   // scan workspace for scaled_params (P <= 1024)
    const int t = threadIdx.x;

    if (t == 0) {
        long long a = 0;
        for (int m = 0; m < nm; ++m) { s_pf[m] = a; a += (long long)fields[m]; }
        s_pf[nm] = a;
    }
    if (t == 1) {
        long long a = 0;
        for (int f = 0; f < nf; ++f) { s_pp[f] = a; a += (long long)ppf[f]; }
        s_pp[nf] = a;
    }

    // Hillis-Steele inclusive scan of scaled_params, then convert to exclusive.
    long long v = (t < P) ? (long long)sp[t] : 0;
    s_ps[t] = v;
    __syncthreads();
    for (int off = 1; off < P; off <<= 1) {
        long long add = (t >= off) ? s_ps[t - off] : 0;
        __syncthreads();
        s_ps[t] += add;
        __syncthreads();
    }
    long long excl = s_ps[t] - v;
    if (t < P) {
        param_start[t] = excl;
        winner[t] = -1;
    }
    __syncthreads();
    s_ps[t] = excl;          // shared now holds exclusive prefix (ps)
    __syncthreads();

    for (int r = t; r < R; r += blockDim.x) {
        const int c0 = coords[3 * r + 0];
        const int c1 = coords[3 * r + 1];
        const int c2 = coords[3 * r + 2];
        const long long af = s_pf[c0] + c1;          // absolute field
        const long long ap = s_pp[af] + c2;          // absolute param
        starts[r] = s_ps[ap];
        atomicMax(&winner[(int)ap], r);              // sequential-scatter: last r wins
    }
}

// ---------------------------------------------------------------------------
// Kernel 2: gathered[r, :] = db[starts[r] : +blk]  via the async data mover.
// Each block moves 16 KB: KU async b128 loads/thread -> LDS -> KU async NT
// stores. Data never touches VGPRs; db loads stay RT so db remains in L2.
// ---------------------------------------------------------------------------
__global__ void gather_kernel(const float* __restrict__ db,
                              const long long* __restrict__ starts,
                              float* __restrict__ out,
                              int blk, int bpr) {
    __shared__ __align__(16) float4 lbuf[TPB * KU];   // 16 KB
    const int bid = blockIdx.x;
    const int r = bid / bpr;
    const int c = bid % bpr;
    const int t = threadIdx.x;

    const float* src = db + starts[r];
    float* dst = out + (long long)r * blk;
    const int f4base = c * (TPB * KU) + t;            // float4 index within the block

#pragma unroll
    for (int k = 0; k < KU; ++k) {
        const int f4 = f4base + k * TPB;
        async_ld_b128(lds_off(&lbuf[k * TPB + t]),
                      (unsigned long long)(size_t)(src + (long long)f4 * 4));
    }
    wait_async0();                                    // all this wave's loads landed
#pragma unroll
    for (int k = 0; k < KU; ++k) {
        const int f4 = f4base + k * TPB;
        async_st_b128_nt((unsigned long long)(size_t)(dst + (long long)f4 * 4),
                         lds_off(&lbuf[k * TPB + t]));
    }
    // S_ENDPGM performs an implicit wait-idle, draining remaining ASYNCcnt.
}

// ---------------------------------------------------------------------------
// Kernel 3: new_db block p = results[winner[p]] if hit else db block p.
// NT b128 loads (read-once sources) and NT b128 stores (write-once dest).
// ---------------------------------------------------------------------------
__global__ void newdb_kernel(const float* __restrict__ db,
                             const float* __restrict__ results,
                             const int* __restrict__ winner,
                             const long long* __restrict__ param_start,
                             float* __restrict__ out,
                             int blk, int bpr) {
    const int bid = blockIdx.x;
    const int p = bid / bpr;
    const int c = bid % bpr;
    const int t = threadIdx.x;

    const long long base = param_start[p];
    const int w = winner[p];
    const v4f* __restrict__ src = (w >= 0)
        ? (const v4f*)(results + (long long)w * blk)
        : (const v4f*)(db + base);
    v4f* __restrict__ dst = (v4f*)(out + base);

    const int f0 = c * (TPB * KU) + t;
#pragma unroll
    for (int k = 0; k < KU; ++k) {
        v4f x = __builtin_nontemporal_load(&src[f0 + k * TPB]);
        __builtin_nontemporal_store(x, &dst[f0 + k * TPB]);
    }
}

// ---------------------------------------------------------------------------
extern "C" void kernel_launch(void* const* d_in, const int* in_sizes, int n_in,
                              void* d_out, int out_size, void* d_ws, size_t ws_size,
                              hipStream_t stream) {
    const float* db      = (const float*)d_in[0];
    const float* results = (const float*)d_in[1];
    const int*   fields  = (const int*)d_in[2];
    const int*   ppf     = (const int*)d_in[3];
    const int*   sp      = (const int*)d_in[4];
    const int*   coords  = (const int*)d_in[5];

    const int nm  = in_sizes[2];          // 8
    const int nf  = in_sizes[3];          // 64
    const int P   = in_sizes[4];          // 512
    const int R   = in_sizes[5] / 3;      // 1024
    const int blk = in_sizes[1] / R;      // 65536

    float* out_g  = (float*)d_out;                         // gathered [R, blk]
    float* out_db = (float*)d_out + (size_t)R * blk;       // new_db   [P*blk]

    long long* param_start = (long long*)d_ws;             // [P]
    long long* starts      = param_start + P;              // [R]
    int*       winner      = (int*)(starts + R);           // [P]

    setup_kernel<<<1, 1024, 0, stream>>>(fields, ppf, sp, coords,
                                         nm, nf, P, R,
                                         param_start, starts, winner);

    const int floats_per_block = TPB * 4 * KU;             // 4096
    const int bpr = blk / floats_per_block;                // 16

    gather_kernel<<<R * bpr, TPB, 0, stream>>>(db, starts, out_g, blk, bpr);
    newdb_kernel<<<P * bpr, TPB, 0, stream>>>(db, results, winner, param_start,
                                              out_db, blk, bpr);
}